// ParallelRNN_4801773437169
// MI455X (gfx1250) — compile-verified
//
#include <hip/hip_runtime.h>
#include <math.h>

// ParallelRNN (LRU): Bx = x @ B^T (V_WMMA_F32_16X16X4_F32), then complex
// recurrence out[t] = lam*out[t-1] + Bx[t], out[-1] = 1 (h0), as a chunked
// wave-parallel affine scan. Output = [real | imag] on last axis.
// BATCH=16, SEQ=1024, D_IN=128, H=64. Memory-bound: ~24 MB total HBM traffic.

#define BATCHN 16
#define SEQN   1024
#define DINN   128
#define HN     64
#define LDB    132   // padded LDS stride (floats): conflict-free WMMA frag reads

typedef __attribute__((ext_vector_type(2))) float v2f;
typedef __attribute__((ext_vector_type(8))) float v8f;

// ---------------------------------------------------------------------------
// Kernel 1: bxT[b, h, t] = sum_j x[b,t,j] * B[h,j]   (transposed output!)
// Grid: 128 blocks x 256 threads (8 waves). Block covers 128 consecutive t of
// one batch; each wave one 16-row tile x all 64 h (4 accumulators, 128 WMMAs).
// After the K loop the block transposes its 128t x 64h result through LDS
// (reusing the B buffer) so the global store is fully coalesced b128s.
// ---------------------------------------------------------------------------
__global__ __launch_bounds__(256)
void gemm_bx_wmma(const float* __restrict__ x, const float* __restrict__ Bm,
                  float* __restrict__ bxT) {
  __shared__ float lds[HN * LDB];   // phase 1: B matrix; phase 2: D staging

  // Stage B (64x128 f32 = 32 KB) into LDS, coalesced.
  for (int idx = threadIdx.x; idx < HN * DINN; idx += 256) {
    int h = idx >> 7, j = idx & 127;
    lds[h * LDB + j] = Bm[idx];
  }
  __syncthreads();

  const int wave = threadIdx.x >> 5;
  const int lane = threadIdx.x & 31;
  const int m    = lane & 15;   // A row within tile / B-D column within tile
  const int hi   = lane >> 4;   // lane half: selects K pair (A,B), +8 rows (D)
  const int rowBase = blockIdx.x * 128 + wave * 16;   // global row (b*SEQ + t)

  // A frag (ISA 16x4 f32 layout): lane holds x[rowBase+m][k + 2*hi + {0,1}]
  const float* xr = x + (size_t)(rowBase + m) * DINN + 2 * hi;
  // B frag (B^T, 4x16 K-major): lane holds B[hTile+m][k + 2*hi + {0,1}]
  const float* bl = &lds[m * LDB + 2 * hi];

  v8f acc0 = {}, acc1 = {}, acc2 = {}, acc3 = {};
  for (int k = 0; k < DINN; k += 4) {
    v2f a   = *(const v2f*)(xr + k);
    v2f bb0 = *(const v2f*)(bl + 0 * 16 * LDB + k);
    v2f bb1 = *(const v2f*)(bl + 1 * 16 * LDB + k);
    v2f bb2 = *(const v2f*)(bl + 2 * 16 * LDB + k);
    v2f bb3 = *(const v2f*)(bl + 3 * 16 * LDB + k);
    acc0 = __builtin_amdgcn_wmma_f32_16x16x4_f32(false, a, false, bb0, (short)0, acc0, false, false);
    acc1 = __builtin_amdgcn_wmma_f32_16x16x4_f32(false, a, false, bb1, (short)0, acc1, false, false);
    acc2 = __builtin_amdgcn_wmma_f32_16x16x4_f32(false, a, false, bb2, (short)0, acc2, false, false);
    acc3 = __builtin_amdgcn_wmma_f32_16x16x4_f32(false, a, false, bb3, (short)0, acc3, false, false);
  }
  __syncthreads();   // all waves done reading B from LDS

  // Stage D into LDS as [h][t_local] (t_local in [0,128)).
  // D layout: VGPR r holds row (t = 8*hi + r), column N = m.
  const int tl = wave * 16 + 8 * hi;
  for (int r = 0; r < 8; ++r) {
    lds[(0 * 16 + m) * LDB + tl + r] = acc0[r];
    lds[(1 * 16 + m) * LDB + tl + r] = acc1[r];
    lds[(2 * 16 + m) * LDB + tl + r] = acc2[r];
    lds[(3 * 16 + m) * LDB + tl + r] = acc3[r];
  }
  __syncthreads();

  // Coalesced transposed store: bxT[(b*64 + h)*1024 + tb0 + t_local]
  const int bB  = blockIdx.x >> 3;           // batch of this block
  const int tb0 = (blockIdx.x & 7) * 128;    // t base of this block
  for (int idx = threadIdx.x; idx < HN * 32; idx += 256) {
    int h = idx >> 5, t4 = (idx & 31) * 4;
    float4 v = *(const float4*)(&lds[h * LDB + t4]);
    *(float4*)(&bxT[((size_t)bB * HN + h) * SEQN + tb0 + t4]) = v;
  }
}

// ---------------------------------------------------------------------------
// Kernel 2: per (b,h): out[t] = lam*out[t-1] + bx[t], out[-1] = 1 (h0 term).
// One wave per (b,h): 32 lanes x 32-step chunks, chunk values kept in
// registers (single pass over bxT, b128 loads). Carries via affine-compose
// shuffle scan: f_c(x) = lam^32 * x + y_last_c (complex, closed-form lam^32).
// Grid: 128 blocks x 256 threads (8 waves) = 1024 (b,h) pairs.
// ---------------------------------------------------------------------------
__global__ __launch_bounds__(256)
void lru_scan(const float* __restrict__ bxT, const float* __restrict__ nu,
              const float* __restrict__ theta, float* __restrict__ out) {
  const int wave = threadIdx.x >> 5;
  const int lane = threadIdx.x & 31;
  const int id   = blockIdx.x * 8 + wave;    // flat (b,h)
  const int b    = id >> 6;
  const int h    = id & 63;

  const float a   = __expf(nu[h]);
  const float th  = theta[h];
  const float mag = __expf(-a);
  const float lre = mag * __cosf(th);
  const float lim = mag * __sinf(th);
  const float m32 = __expf(-32.0f * a);
  float p_re = m32 * __cosf(32.0f * th);
  float p_im = m32 * __sinf(32.0f * th);

  // Chunk-local recurrence (zero carry); keep all 32 values in registers.
  const float4* u4 = (const float4*)(bxT + (size_t)id * SEQN + lane * 32);
  float yl_re[32], yl_im[32];
  float yre = 0.0f, yim = 0.0f;
#pragma unroll
  for (int j = 0; j < 8; ++j) {
    const float4 uu = u4[j];
    const float us[4] = {uu.x, uu.y, uu.z, uu.w};
#pragma unroll
    for (int e = 0; e < 4; ++e) {
      const float nre = lre * yre - lim * yim + us[e];
      const float nim = lre * yim + lim * yre;
      yre = nre; yim = nim;
      yl_re[j * 4 + e] = yre;
      yl_im[j * 4 + e] = yim;
    }
  }
  float q_re = yre, q_im = yim;

  // Inclusive scan of affine maps f(x) = p*x + q (complex), 5 shuffle steps.
#pragma unroll
  for (int d = 1; d < 32; d <<= 1) {
    const float ppr = __shfl_up(p_re, d, 32);
    const float ppi = __shfl_up(p_im, d, 32);
    const float qqr = __shfl_up(q_re, d, 32);
    const float qqi = __shfl_up(q_im, d, 32);
    if (lane >= d) {
      const float nqr = p_re * qqr - p_im * qqi + q_re;   // q <- p*qq + q
      const float nqi = p_re * qqi + p_im * qqr + q_im;
      const float npr = p_re * ppr - p_im * ppi;          // p <- p*pp
      const float npi = p_re * ppi + p_im * ppr;
      q_re = nqr; q_im = nqi; p_re = npr; p_im = npi;
    }
  }

  // Exclusive carry applied to initial state h0 = 1: carry = p_ex*1 + q_ex.
  float cer = __shfl_up(p_re, 1, 32) + __shfl_up(q_re, 1, 32);
  float cei = __shfl_up(p_im, 1, 32) + __shfl_up(q_im, 1, 32);
  if (lane == 0) { cer = 1.0f; cei = 0.0f; }

  // out[t0+i] = y_local[i] + lam^(i+1) * carry ; write [real | imag].
  float* o = out + ((size_t)b * SEQN + (size_t)lane * 32) * (2 * HN) + h;
  float pwre = lre, pwim = lim;
#pragma unroll
  for (int i = 0; i < 32; ++i) {
    const float ore = yl_re[i] + pwre * cer - pwim * cei;
    const float oim = yl_im[i] + pwre * cei + pwim * cer;
    o[(size_t)i * (2 * HN)]      = ore;
    o[(size_t)i * (2 * HN) + HN] = oim;
    const float npr = pwre * lre - pwim * lim;   // pw *= lam
    const float npi = pwre * lim + pwim * lre;
    pwre = npr; pwim = npi;
  }
}

extern "C" void kernel_launch(void* const* d_in, const int* in_sizes, int n_in,
                              void* d_out, int out_size, void* d_ws, size_t ws_size,
                              hipStream_t stream) {
  (void)in_sizes; (void)n_in; (void)out_size; (void)ws_size;
  const float* x  = (const float*)d_in[0];   // [16,1024,128]
  const float* Bm = (const float*)d_in[1];   // [64,128]
  const float* nu = (const float*)d_in[2];   // [64]
  const float* th = (const float*)d_in[3];   // [64]
  float* out = (float*)d_out;                // [16,1024,128]
  float* bxT = (float*)d_ws;                 // [16,64,1024] f32 = 4 MB scratch

  gemm_bx_wmma<<<128, 256, 0, stream>>>(x, Bm, bxT);
  lru_scan<<<128, 256, 0, stream>>>(bxT, nu, th, out);
}